// AnchorFreePS_472446402879
// MI455X (gfx1250) — compile-verified
//
#include <hip/hip_runtime.h>
#include <math.h>

#define W_IMG 1280.0f
#define H_IMG 1280.0f
#define N_POINTS 34100
#define BATCH 8
#define N_GT 200
#define NCAND 3500
#define MAX_DET 300
#define NMS_THRESH 0.5f
#define SCORE_THRESH 0.05f
#define NEG_INF (-__builtin_inff())

typedef __attribute__((ext_vector_type(2))) float v2f;
typedef __attribute__((ext_vector_type(8))) float v8f;

// ---------------------------------------------------------------------------
// Kernel 1: point-in-box matching. gt boxes for this image staged into LDS via
// CDNA5 async global->LDS copies (ASYNCcnt path), then register loop over 200.
// ---------------------------------------------------------------------------
__global__ void match_kernel(const float* __restrict__ points,
                             const float* __restrict__ gt,    // [8][200][4]
                             int* __restrict__ matched) {     // [8][N_POINTS]
    __shared__ float sb[N_GT * 4];
    const int b   = blockIdx.y;
    const int tid = threadIdx.x;

    if (tid < N_GT) {
        unsigned loff = (unsigned)(size_t)&sb[tid * 4];             // LDS byte offset
        unsigned long long ga =
            (unsigned long long)(const void*)(gt + ((size_t)b * N_GT + tid) * 4);
        asm volatile("global_load_async_to_lds_b128 %0, %1, off"
                     :: "v"(loff), "v"(ga) : "memory");
    }
    asm volatile("s_wait_asynccnt 0x0" ::: "memory");
    __syncthreads();

    const int p = blockIdx.x * blockDim.x + tid;
    if (p >= N_POINTS) return;

    const float px = points[2 * p + 0];
    const float py = points[2 * p + 1];
    const float BIG = 2147483647.0f;

    float bestc = 3.4e38f;
    int   besti = -1;
#pragma unroll 4
    for (int g = 0; g < N_GT; ++g) {
        float x1 = sb[g * 4 + 0], y1 = sb[g * 4 + 1];
        float x2 = sb[g * 4 + 2], y2 = sb[g * 4 + 3];
        bool inside = (px >= x1) & (px <= x2) & (py >= y1) & (py <= y2);
        float area  = (x2 - x1) * (y2 - y1);
        float cost  = inside ? area : BIG;
        if (cost < bestc) { bestc = cost; besti = g; }
    }
    matched[(size_t)b * N_POINTS + p] = (bestc < BIG) ? besti : -1;
}

// ---------------------------------------------------------------------------
// Kernel 2: decode+clip ALL boxes via V_WMMA_F32_16X16X4_F32.
//   D(16x4) = A1(16x4 [x,y,0,0]) * B1  +  A2(16x4 [l,t,r,b]) * B2
// One wave decodes one 16-point tile with two chained WMMAs. Results are
// clipped, staged in a per-wave LDS tile, and written out with
// global_store_async_from_lds_b128 (one contiguous 16B store per box).
// ---------------------------------------------------------------------------
__global__ void decode_kernel(const float* __restrict__ points,
                              const float* __restrict__ regression, // [8][N][4]
                              float* __restrict__ dec) {            // [8][N][4]
    __shared__ float stage[8][64];          // per-wave 16 boxes x 4 coords
    const int tiles_per_img = (N_POINTS + 15) / 16;                  // 2132
    const int lane = threadIdx.x & 31;
    const int wave = threadIdx.x >> 5;
    const int tile = blockIdx.x * (blockDim.x >> 5) + wave;
    if (tile >= BATCH * tiles_per_img) return;                       // wave-uniform

    const int b  = tile / tiles_per_img;
    const int t  = tile - b * tiles_per_img;
    const int p0 = t * 16;
    const int m  = lane & 15;
    const bool hiK = lane >= 16;

    int p = p0 + m;
    if (p >= N_POINTS) p = N_POINTS - 1;                             // clamp tail loads

    const float px = points[2 * p + 0];
    const float py = points[2 * p + 1];
    const float* rg = regression + (((size_t)b * N_POINTS) + p) * 4;

    // A layout (16x4 f32): lanes 0-15 hold K=0,1 in v0,v1; lanes 16-31 hold K=2,3.
    v2f a1, a2, b1, b2;
    a1.x = hiK ? 0.f : px;      a1.y = hiK ? 0.f : py;               // [x,y,0,0]
    a2.x = hiK ? rg[2] : rg[0]; a2.y = hiK ? rg[3] : rg[1];          // [l,t,r,b]
    // B layout (4x16): v0 = rows K=0 (lanes 0-15) / K=2 (lanes 16-31); v1 = K=1 / K=3.
    b1.x = (!hiK && (m == 0 || m == 2)) ? 1.f : 0.f;                 // x -> x1,x2
    b1.y = (!hiK && (m == 1 || m == 3)) ? 1.f : 0.f;                 // y -> y1,y2
    b2.x = hiK ? ((m == 2) ?  W_IMG : 0.f) : ((m == 0) ? -W_IMG : 0.f);
    b2.y = hiK ? ((m == 3) ?  H_IMG : 0.f) : ((m == 1) ? -H_IMG : 0.f);

    v8f c = {};
    c = __builtin_amdgcn_wmma_f32_16x16x4_f32(false, a1, false, b1, (short)0, c, false, false);
    c = __builtin_amdgcn_wmma_f32_16x16x4_f32(false, a2, false, b2, (short)0, c, false, false);

    // D layout: VGPR r -> M=r (lanes 0-15) / M=8+r (lanes 16-31), N=lane%16.
    // Clip and stage row-major per point into this wave's LDS tile.
    if (m < 4) {
        const float lim = 1280.f;            // W_IMG == H_IMG
#pragma unroll
        for (int r = 0; r < 8; ++r) {
            const int row = (hiK ? 8 : 0) + r;
            stage[wave][row * 4 + m] = fminf(fmaxf(c[r], 0.f), lim);
        }
    }
    asm volatile("s_wait_dscnt 0x0" ::: "memory");   // LDS writes visible

    // one contiguous async b128 store per decoded box
    if (lane < 16) {
        const int pr = p0 + lane;
        if (pr < N_POINTS) {
            unsigned loff = (unsigned)(size_t)&stage[wave][lane * 4];
            unsigned long long ga =
                (unsigned long long)(void*)(dec + (((size_t)b * N_POINTS) + pr) * 4);
            asm volatile("global_store_async_from_lds_b128 %0, %1, off"
                         :: "v"(ga), "v"(loff) : "memory");
        }
    }
    asm volatile("s_wait_asynccnt 0x0" ::: "memory");
}

// ---------------------------------------------------------------------------
// Kernel 3: per-(batch,level) top-k via 4096-bin LDS radix histogram on the
// order-preserving bit transform of the raw logit (sigmoid is monotone).
// Exactly k candidates are emitted per level; score = sigmoid(logit).
// ---------------------------------------------------------------------------
__device__ __forceinline__ unsigned ordkey(float x) {
    unsigned u = __float_as_uint(x);
    return (u & 0x80000000u) ? ~u : (u | 0x80000000u);
}

__global__ void topk_kernel(const float* __restrict__ logits,     // [8][N][1]
                            float* __restrict__ cand_score,       // [8][3500]
                            int* __restrict__ cand_idx) {         // [8][3500]
    const int lvl_off[5]  = {0, 25600, 32000, 33600, 34000};
    const int lvl_size[5] = {25600, 6400, 1600, 400, 100};
    const int lvl_k[5]    = {1000, 1000, 1000, 400, 100};
    const int lvl_co[5]   = {0, 1000, 2000, 3000, 3400};

    const int b  = blockIdx.x;
    const int lv = blockIdx.y;
    const int off0 = lvl_off[lv], n = lvl_size[lv];
    const int K = lvl_k[lv], co = lvl_co[lv];
    const float* lg = logits + (size_t)b * N_POINTS + off0;
    const int tid = threadIdx.x;

    __shared__ unsigned hist[4096];
    __shared__ unsigned s_T, s_fill;

    for (int i = tid; i < 4096; i += 1024) hist[i] = 0;
    __syncthreads();

    for (int i = tid; i < n; i += 1024)
        atomicAdd(&hist[ordkey(lg[i]) >> 20], 1u);
    __syncthreads();

    if (tid == 0) {
        unsigned acc = 0; int T = 0;
        for (int bin = 4095; bin >= 0; --bin) {
            unsigned nb = acc + hist[bin];
            if (nb >= (unsigned)K) { T = bin; break; }
            acc = nb;
        }
        s_T = (unsigned)T; s_fill = 0;
    }
    __syncthreads();
    const unsigned T = s_T;

    // pass 2a: strictly-above-threshold bins (count < K by construction)
    for (int i = tid; i < n; i += 1024) {
        float x = lg[i];
        if ((ordkey(x) >> 20) > T) {
            unsigned pos = atomicAdd(&s_fill, 1u);
            cand_score[(size_t)b * NCAND + co + pos] = 1.f / (1.f + __expf(-x));
            cand_idx  [(size_t)b * NCAND + co + pos] = off0 + i;
        }
    }
    __syncthreads();
    // pass 2b: threshold bin fills the remaining slots up to exactly K
    for (int i = tid; i < n; i += 1024) {
        float x = lg[i];
        if ((ordkey(x) >> 20) == T) {
            unsigned pos = atomicAdd(&s_fill, 1u);
            if (pos < (unsigned)K) {
                cand_score[(size_t)b * NCAND + co + pos] = 1.f / (1.f + __expf(-x));
                cand_idx  [(size_t)b * NCAND + co + pos] = off0 + i;
            }
        }
    }
}

// ---------------------------------------------------------------------------
// Kernel 4: greedy NMS. One workgroup per image; scores live in LDS, each
// thread's <=7 candidate boxes cached in registers; argmax = wave32 shfl_xor
// reduction + one LDS hop across the 16 waves.
// ---------------------------------------------------------------------------
#define NMS_THREADS 512
#define CPT 7   // ceil(3500/512)

__global__ void nms_kernel(const float* __restrict__ cand_score,
                           const int* __restrict__ cand_idx,
                           const float* __restrict__ dec,
                           float* __restrict__ out_boxes,   // [8][300][4]
                           float* __restrict__ out_scores,  // [8][300]
                           int* __restrict__ out_labels,    // [8][300]
                           int* __restrict__ out_valid) {   // [8][300]
    const int b = blockIdx.x;
    const int tid = threadIdx.x;

    __shared__ float work[NCAND];
    __shared__ float sbox[4];
    __shared__ float s_s;
    __shared__ float red_s[NMS_THREADS / 32];
    __shared__ int   red_i[NMS_THREADS / 32];

    float bx[CPT][4], area[CPT];
#pragma unroll
    for (int k = 0; k < CPT; ++k) {
        int j = tid + k * NMS_THREADS;
        if (j < NCAND) {
            int pi = cand_idx[(size_t)b * NCAND + j];
            const float* d = &dec[(((size_t)b * N_POINTS) + pi) * 4];
            bx[k][0] = d[0]; bx[k][1] = d[1]; bx[k][2] = d[2]; bx[k][3] = d[3];
            area[k]  = (bx[k][2] - bx[k][0]) * (bx[k][3] - bx[k][1]);
            float s = cand_score[(size_t)b * NCAND + j];
            work[j] = (s > SCORE_THRESH) ? s : NEG_INF;
        }
    }
    __syncthreads();

    for (int it = 0; it < MAX_DET; ++it) {
        // ---- block argmax over work[] ----
        float best = NEG_INF; int bi = -1;
#pragma unroll
        for (int k = 0; k < CPT; ++k) {
            int j = tid + k * NMS_THREADS;
            if (j < NCAND) {
                float w = work[j];
                if (w > best || (w == best && bi >= 0 && j < bi)) { best = w; bi = j; }
                if (bi < 0 && w > NEG_INF) { best = w; bi = j; }
            }
        }
#pragma unroll
        for (int off = 16; off > 0; off >>= 1) {
            float os = __shfl_xor(best, off, 32);
            int   oi = __shfl_xor(bi, off, 32);
            if (os > best || (os == best && oi >= 0 && (unsigned)oi < (unsigned)bi)) {
                best = os; bi = oi;
            }
        }
        const int wv = tid >> 5;
        if ((tid & 31) == 0) { red_s[wv] = best; red_i[wv] = bi; }
        __syncthreads();
        if (tid == 0) {
            for (int w = 1; w < NMS_THREADS / 32; ++w) {
                if (red_s[w] > best ||
                    (red_s[w] == best && red_i[w] >= 0 && (unsigned)red_i[w] < (unsigned)bi)) {
                    best = red_s[w]; bi = red_i[w];
                }
            }
            s_s = best;
            bool valid = (best > NEG_INF) && (bi >= 0);
            float bb[4] = {0.f, 0.f, 0.f, 0.f};
            if (valid) {
                int pi = cand_idx[(size_t)b * NCAND + bi];
                const float* d = &dec[(((size_t)b * N_POINTS) + pi) * 4];
                bb[0] = d[0]; bb[1] = d[1]; bb[2] = d[2]; bb[3] = d[3];
            }
            sbox[0] = bb[0]; sbox[1] = bb[1]; sbox[2] = bb[2]; sbox[3] = bb[3];
            const size_t o = (size_t)b * MAX_DET + it;
            out_boxes[o * 4 + 0] = bb[0]; out_boxes[o * 4 + 1] = bb[1];
            out_boxes[o * 4 + 2] = bb[2]; out_boxes[o * 4 + 3] = bb[3];
            out_scores[o] = valid ? best : 0.f;
            out_labels[o] = valid ? 0 : -1;
            out_valid[o]  = valid ? 1 : 0;
        }
        __syncthreads();

        // ---- suppress vs picked box ----
        if (s_s > NEG_INF) {
            const float x1 = sbox[0], y1 = sbox[1], x2 = sbox[2], y2 = sbox[3];
            const float a1 = (x2 - x1) * (y2 - y1);
#pragma unroll
            for (int k = 0; k < CPT; ++k) {
                int j = tid + k * NMS_THREADS;
                if (j < NCAND && work[j] > NEG_INF) {
                    float xx1 = fmaxf(x1, bx[k][0]);
                    float yy1 = fmaxf(y1, bx[k][1]);
                    float xx2 = fminf(x2, bx[k][2]);
                    float yy2 = fminf(y2, bx[k][3]);
                    float inter = fmaxf(xx2 - xx1, 0.f) * fmaxf(yy2 - yy1, 0.f);
                    float iou = inter / (a1 + area[k] - inter + 1e-9f);
                    if (iou > NMS_THRESH) work[j] = NEG_INF;
                }
            }
        }
        __syncthreads();
    }
}

// ---------------------------------------------------------------------------
// launch
// ---------------------------------------------------------------------------
extern "C" void kernel_launch(void* const* d_in, const int* in_sizes, int n_in,
                              void* d_out, int out_size, void* d_ws, size_t ws_size,
                              hipStream_t stream) {
    (void)in_sizes; (void)n_in; (void)out_size; (void)ws_size;
    const float* points     = (const float*)d_in[0];   // [N_POINTS][2]
    const float* gt_boxes   = (const float*)d_in[1];   // [8][200][4]
    const float* cls_logits = (const float*)d_in[2];   // [8][N_POINTS][1]
    const float* regression = (const float*)d_in[3];   // [8][N_POINTS][4]

    // workspace layout
    char* ws = (char*)d_ws;
    float* dec = (float*)ws;                                         // 8*N*4 f32
    size_t off = (size_t)BATCH * N_POINTS * 4 * sizeof(float);
    off = (off + 255) & ~(size_t)255;
    float* cand_score = (float*)(ws + off);
    off += (size_t)BATCH * NCAND * sizeof(float);
    off = (off + 255) & ~(size_t)255;
    int* cand_idx = (int*)(ws + off);

    // output layout (flat 4-byte slots, reference return order)
    int*   out_i       = (int*)d_out;
    float* out_f       = (float*)d_out;
    int*   matched     = out_i;                                // [0, 272800)
    float* out_boxes   = out_f + BATCH * N_POINTS;             // +9600
    float* out_scores  = out_boxes + BATCH * MAX_DET * 4;      // +2400
    int*   out_labels  = (int*)(out_scores + BATCH * MAX_DET); // +2400
    int*   out_valid   = out_labels + BATCH * MAX_DET;         // +2400

    dim3 mgrid((N_POINTS + 255) / 256, BATCH);
    match_kernel<<<mgrid, 256, 0, stream>>>(points, gt_boxes, matched);

    const int tiles = BATCH * ((N_POINTS + 15) / 16);          // 17056
    decode_kernel<<<(tiles + 7) / 8, 256, 0, stream>>>(points, regression, dec);

    topk_kernel<<<dim3(BATCH, 5), 1024, 0, stream>>>(cls_logits, cand_score, cand_idx);

    nms_kernel<<<BATCH, NMS_THREADS, 0, stream>>>(cand_score, cand_idx, dec,
                                                  out_boxes, out_scores,
                                                  out_labels, out_valid);
}